// AttentionFlow_20598663152052
// MI455X (gfx1250) — compile-verified
//
#include <hip/hip_runtime.h>
#include <hip/hip_bf16.h>
#include <math.h>

// ---------------------------------------------------------------- constants
#define BATCH 64
#define EPB   1024            // edges per batch
#define NEDGE (BATCH * EPB)   // 65536
#define MNODE 65536
#define DDIM  128
#define D4    512
#define TOPK  64
#define LDSTRIDE 132          // 16-row X tile, padded to avoid LDS bank conflicts

typedef float v2f __attribute__((ext_vector_type(2)));
typedef float v8f __attribute__((ext_vector_type(8)));

__device__ __forceinline__ float dot4(float4 a, float4 b) {
    return a.x * b.x + a.y * b.y + a.z * b.z + a.w * b.w;
}
// order-preserving float<->uint key for atomicMax-based segment max
__device__ __forceinline__ unsigned fkey(float f) {
    unsigned u = __float_as_uint(f);
    return (u & 0x80000000u) ? ~u : (u | 0x80000000u);
}
__device__ __forceinline__ float funkey(unsigned k) {
    unsigned u = (k & 0x80000000u) ? (k & 0x7FFFFFFFu) : ~k;
    return __uint_as_float(u);
}

// ---------------------------------------------------------------- K0: Mqk = Wq^T @ Wk  (512x512)
// One 16x16 C tile per wave via V_WMMA_F32_16X16X4_F32; A = Wq^T (strided
// reads are lane-coalesced since consecutive lanes read consecutive columns).
__global__ __launch_bounds__(128) void mqk_kernel(const float* __restrict__ Wq,
                                                  const float* __restrict__ Wk,
                                                  float* __restrict__ Mqk) {
    const int row0 = blockIdx.x * 16;
    const int wave = threadIdx.x >> 5;
    const int col0 = (blockIdx.y * 4 + wave) * 16;
    const int lane = threadIdx.x & 31;
    const int mn = lane & 15;        // row of A / col of B for this lane
    const int kh = (lane >> 4) * 2;  // K sub-offset: lanes 16-31 hold K+2,K+3
    v8f acc = {0.f, 0.f, 0.f, 0.f, 0.f, 0.f, 0.f, 0.f};
    const float* ap = Wq + row0 + mn;   // A[m,k] = Wq[k, row0+m]
    const float* bp = Wk + col0 + mn;   // B[k,n] = Wk[k, col0+n]
    for (int k = 0; k < D4; k += 4) {
        v2f a, b;
        a.x = ap[(k + kh) * D4];
        a.y = ap[(k + kh + 1) * D4];
        b.x = bp[(k + kh) * D4];
        b.y = bp[(k + kh + 1) * D4];
        acc = __builtin_amdgcn_wmma_f32_16x16x4_f32(false, a, false, b,
                                                    (short)0, acc, false, false);
    }
#pragma unroll
    for (int v = 0; v < 8; ++v)  // VGPR v holds rows {v, v+8}
        Mqk[(row0 + v + (kh << 2)) * D4 + col0 + mn] = acc[v];
}

// ---------------------------------------------------------------- K0b: W3 = M10^T (128x128)
__global__ void trans_kernel(const float* __restrict__ Mqk, float* __restrict__ W3) {
    int t = blockIdx.x * 256 + threadIdx.x;   // 16384 elements
    int k = t >> 7, d = t & 127;
    W3[t] = Mqk[(128 + d) * D4 + k];          // W3[k,d] = M[128+d, k]
}

// ---------------------------------------------------------------- K1: C[R x Ncols] = X[R x 128] @ W
// X tile (16x128) staged in LDS once per block, 4 waves cover 64 output cols.
__global__ __launch_bounds__(128) void gemm_xw(const float* __restrict__ X,
                                               const float* __restrict__ W, int ldw,
                                               float* __restrict__ C, int ldc) {
    __shared__ float xs[16 * LDSTRIDE];
    const int row0 = blockIdx.x * 16;
    const int wave = threadIdx.x >> 5;
    const int col0 = (blockIdx.y * 4 + wave) * 16;
    {   // cooperative coalesced tile load: 16x128 f32 = 512 float4
        const float4* Xv = (const float4*)(X + (size_t)row0 * DDIM);
        for (int i = threadIdx.x; i < 16 * 32; i += 128) {
            int r = i >> 5, c = i & 31;
            float4 v = Xv[r * 32 + c];
            float* dst = &xs[r * LDSTRIDE + c * 4];
            dst[0] = v.x; dst[1] = v.y; dst[2] = v.z; dst[3] = v.w;
        }
    }
    __syncthreads();
    const int lane = threadIdx.x & 31;
    const int mn = lane & 15;
    const int kh = (lane >> 4) * 2;
    v8f acc = {0.f, 0.f, 0.f, 0.f, 0.f, 0.f, 0.f, 0.f};
    const float* wp = W + col0 + mn;
#pragma unroll 4
    for (int k = 0; k < DDIM; k += 4) {
        v2f a, b;
        a.x = xs[mn * LDSTRIDE + k + kh];
        a.y = xs[mn * LDSTRIDE + k + kh + 1];
        b.x = wp[(size_t)(k + kh) * ldw];
        b.y = wp[(size_t)(k + kh + 1) * ldw];
        acc = __builtin_amdgcn_wmma_f32_16x16x4_f32(false, a, false, b,
                                                    (short)0, acc, false, false);
    }
    float* cp = C + col0 + mn;
#pragma unroll
    for (int v = 0; v < 8; ++v)
        cp[(size_t)(row0 + v + (kh << 2)) * ldc] = acc[v];
}

// ---------------------------------------------------------------- K2: per-batch vectors & scalar
__global__ __launch_bounds__(128) void batch_kernel(const float* __restrict__ qs_all,
                                                    const float* __restrict__ qr_all,
                                                    const float* __restrict__ M,
                                                    float* __restrict__ w1,
                                                    float* __restrict__ w3,
                                                    float* __restrict__ w24,
                                                    float* __restrict__ cb) {
    __shared__ float z[256];
    __shared__ float red[128];
    const int b = blockIdx.x, d = threadIdx.x;
    z[d] = qs_all[b * DDIM + d];
    z[d + 128] = qr_all[b * DDIM + d];
    __syncthreads();
    float s1 = 0.f, s3 = 0.f, s2 = 0.f, s4 = 0.f;
    for (int k = 0; k < DDIM; ++k) {
        float zq = z[k], zr = z[128 + k];
        s1 += M[d * D4 + 256 + k] * zq + M[d * D4 + 384 + k] * zr;            // M02 qs + M03 qr
        s3 += M[(256 + k) * D4 + d] * zq + M[(384 + k) * D4 + d] * zr;        // M20^T qs + M30^T qr
        s2 += M[(128 + d) * D4 + 256 + k] * zq + M[(128 + d) * D4 + 384 + k] * zr;  // M12/M13
        s4 += M[(256 + k) * D4 + 128 + d] * zq + M[(384 + k) * D4 + 128 + d] * zr;  // M21^T/M31^T
    }
    w1[b * DDIM + d] = s1;
    w3[b * DDIM + d] = s3;
    w24[b * DDIM + d] = s2 + s4;
    // c_b = z^T M[256:512,256:512] z
    float part = 0.f;
#pragma unroll
    for (int ii = 0; ii < 2; ++ii) {
        int i = d + ii * 128;
        float acc = 0.f;
        for (int jj = 0; jj < 256; ++jj) acc += M[(256 + i) * D4 + 256 + jj] * z[jj];
        part += z[i] * acc;
    }
    red[d] = part;
    __syncthreads();
    for (int st = 64; st >= 1; st >>= 1) {
        if (d < st) red[d] += red[d + st];
        __syncthreads();
    }
    if (d == 0) cb[b] = red[0];
}

// ---------------------------------------------------------------- init: segment max keys / sums
__global__ void init_kernel(unsigned* __restrict__ smax, float* __restrict__ den) {
    int i = blockIdx.x * 256 + threadIdx.x;
    smax[i] = 0u;
    den[i] = 0.f;
}

// ---------------------------------------------------------------- K3: per-edge logits (one wave/edge)
__global__ __launch_bounds__(256) void edge_kernel(
        const float* __restrict__ V, const float* __restrict__ rel,
        const int* __restrict__ edges,
        const float* __restrict__ C01, const float* __restrict__ Bj,
        const float* __restrict__ Cr,
        const float* __restrict__ w1, const float* __restrict__ w3,
        const float* __restrict__ w24, const float* __restrict__ cb,
        float* __restrict__ logits, unsigned* __restrict__ smax) {
    const int n = blockIdx.x * 8 + (threadIdx.x >> 5);
    const int lane = threadIdx.x & 31;
    const int b = n >> 10;
    const int s = edges[n * 8 + 6];
    const int j = edges[n * 8 + 7];
    const float4* vs  = (const float4*)(V + (size_t)s * DDIM);
    const float4* vj  = (const float4*)(V + (size_t)j * DDIM);
    const float4* rl  = (const float4*)(rel + (size_t)n * DDIM);
    const float4* a0  = (const float4*)(C01 + (size_t)s * 256);
    const float4* a1  = a0 + 32;
    const float4* bj  = (const float4*)(Bj + (size_t)j * DDIM);
    const float4* cr  = (const float4*)(Cr + (size_t)n * DDIM);
    const float4* w1p = (const float4*)(w1 + b * DDIM);
    const float4* w3p = (const float4*)(w3 + b * DDIM);
    const float4* w2p = (const float4*)(w24 + b * DDIM);
    float4 xvs = vs[lane], xvj = vj[lane], xr = rl[lane];
    float4 xa0 = a0[lane], xa1 = a1[lane], xb = bj[lane], xc = cr[lane];
    float4 x1 = w1p[lane], x3 = w3p[lane], x24 = w2p[lane];
    float sum = dot4(xa0, xvj) + dot4(xa1, xr) + dot4(xvs, x1) + dot4(xr, xb) +
                dot4(xc, xr) + dot4(xr, x24) + dot4(xvj, x3);
    for (int off = 16; off >= 1; off >>= 1) sum += __shfl_xor(sum, off, 32);
    if (lane == 0) {
        float L = sum + cb[b];
        logits[n] = L;
        atomicMax(&smax[s], fkey(L));   // segment max via ordered-uint key
    }
}

// ---------------------------------------------------------------- K4: exp + segment sum
__global__ void expsum_kernel(const int* __restrict__ edges,
                              const float* __restrict__ logits,
                              const unsigned* __restrict__ smax,
                              float* __restrict__ ex, float* __restrict__ den) {
    int n = blockIdx.x * 256 + threadIdx.x;
    int s = edges[n * 8 + 6];
    float e = expf(logits[n] - funkey(smax[s]));
    ex[n] = e;
    atomicAdd(&den[s], e);
}

// ---------------------------------------------------------------- K5: normalize + per-batch top-64 + gathers
__global__ __launch_bounds__(256) void topk_kernel(const int* __restrict__ edges,
                                                   const float* __restrict__ ex,
                                                   const float* __restrict__ den,
                                                   const float* __restrict__ score,
                                                   float* __restrict__ out) {
    __shared__ float ts[EPB];
    __shared__ float sf[EPB];
    __shared__ float rv[256];
    __shared__ int   ri[256];
    const int b = blockIdx.x;
    for (int e = threadIdx.x; e < EPB; e += 256) {
        int n = b * EPB + e;
        int s = edges[n * 8 + 6];
        float so = ex[n] / den[s];
        sf[e] = so;
        ts[e] = so * score[s];
    }
    __syncthreads();
    float* out_score = out;
    float* out_soft  = out + BATCH * TOPK;
    float* out_edges = out + 2 * BATCH * TOPK;
    float* out_orig  = out + 2 * BATCH * TOPK + BATCH * TOPK * 8;
    for (int k = 0; k < TOPK; ++k) {
        float bv = -INFINITY;
        int bi = EPB;
        for (int e = threadIdx.x; e < EPB; e += 256) {
            float v = ts[e];
            if (v > bv || (v == bv && e < bi)) { bv = v; bi = e; }
        }
        rv[threadIdx.x] = bv;
        ri[threadIdx.x] = bi;
        __syncthreads();
        for (int st = 128; st >= 1; st >>= 1) {
            if (threadIdx.x < st) {
                float ov = rv[threadIdx.x + st];
                int   oi = ri[threadIdx.x + st];
                if (ov > rv[threadIdx.x] || (ov == rv[threadIdx.x] && oi < ri[threadIdx.x])) {
                    rv[threadIdx.x] = ov;
                    ri[threadIdx.x] = oi;
                }
            }
            __syncthreads();
        }
        if (threadIdx.x == 0) {
            int e = ri[0], n = b * EPB + e, o = b * TOPK + k;
            out_score[o] = rv[0];
            out_soft[o]  = sf[e];
            out_orig[o]  = (float)n;
            for (int c = 0; c < 8; ++c) out_edges[o * 8 + c] = (float)edges[n * 8 + c];
            ts[e] = -INFINITY;
        }
        __syncthreads();
    }
}

// ---------------------------------------------------------------- host side
extern "C" void kernel_launch(void* const* d_in, const int* in_sizes, int n_in,
                              void* d_out, int out_size, void* d_ws, size_t ws_size,
                              hipStream_t stream) {
    const float* score = (const float*)d_in[0];   // (M,)
    const float* V     = (const float*)d_in[1];   // (M,128)
    const float* rel   = (const float*)d_in[2];   // (N,128)
    const float* qs    = (const float*)d_in[3];   // (B,128)
    const float* qr    = (const float*)d_in[4];   // (B,128)
    const float* Wq    = (const float*)d_in[5];   // (512,512)
    const float* Wk    = (const float*)d_in[6];   // (512,512)
    const int*   edges = (const int*)d_in[7];     // (N,8)
    (void)in_sizes; (void)n_in; (void)out_size; (void)ws_size;

    float* ws = (float*)d_ws;
    size_t off = 0;
    float* Mqk = ws + off; off += (size_t)D4 * D4;          // 512x512
    float* W3  = ws + off; off += (size_t)DDIM * DDIM;      // M10^T
    float* C01 = ws + off; off += (size_t)MNODE * 256;      // [V@M00 | V@M01]
    float* Bj  = ws + off; off += (size_t)MNODE * DDIM;     // V@M10^T
    float* Cr  = ws + off; off += (size_t)NEDGE * DDIM;     // rel@M11
    float* w1  = ws + off; off += (size_t)BATCH * DDIM;
    float* w3  = ws + off; off += (size_t)BATCH * DDIM;
    float* w24 = ws + off; off += (size_t)BATCH * DDIM;
    float* cb  = ws + off; off += 64;
    unsigned* smax = (unsigned*)(ws + off); off += MNODE;
    float* den    = ws + off; off += MNODE;
    float* logits = ws + off; off += NEDGE;
    float* ex     = ws + off; off += NEDGE;

    init_kernel<<<MNODE / 256, 256, 0, stream>>>(smax, den);
    mqk_kernel<<<dim3(D4 / 16, D4 / 64), 128, 0, stream>>>(Wq, Wk, Mqk);
    trans_kernel<<<(DDIM * DDIM) / 256, 256, 0, stream>>>(Mqk, W3);
    // C01 = V @ Mqk[0:128, 0:256]   (A0|A1)
    gemm_xw<<<dim3(MNODE / 16, 256 / 64), 128, 0, stream>>>(V, Mqk, D4, C01, 256);
    // Bj = V @ M10^T
    gemm_xw<<<dim3(MNODE / 16, 128 / 64), 128, 0, stream>>>(V, W3, DDIM, Bj, DDIM);
    // Cr = rel @ M11
    gemm_xw<<<dim3(NEDGE / 16, 128 / 64), 128, 0, stream>>>(rel, Mqk + 128 * D4 + 128, D4, Cr, DDIM);
    batch_kernel<<<BATCH, 128, 0, stream>>>(qs, qr, Mqk, w1, w3, w24, cb);
    edge_kernel<<<NEDGE / 8, 256, 0, stream>>>(V, rel, edges, C01, Bj, Cr,
                                               w1, w3, w24, cb, logits, smax);
    expsum_kernel<<<NEDGE / 256, 256, 0, stream>>>(edges, logits, smax, ex, den);
    topk_kernel<<<BATCH, 256, 0, stream>>>(edges, ex, den, score, (float*)d_out);
}